// MedianLocalActivation_506806141062
// MI455X (gfx1250) — compile-verified
//
#include <hip/hip_runtime.h>
#include <stdint.h>

#define N_NODES 10000
#define D_MAX   128
#define BF_DIM  128           // BATCH * F_DIM = 2 * 64
#define WAVES_PER_BLOCK 8
#define BLOCK_THREADS 256
#define VEC16 (N_NODES / 4)   // 2500 16-byte chunks per feature row
#define CHUNKS_PER_THREAD 10  // ceil(2500 / 256), padded so every wave issues 10

static_assert(N_NODES % WAVES_PER_BLOCK == 0, "grid must cover nodes exactly");

// Async-to-LDS builtin takes int4* pointers (vector_size(16) int) in
// addrspace(1) (global src) and addrspace(3) (LDS dst).
typedef int v4i __attribute__((vector_size(16)));
typedef __attribute__((address_space(1))) v4i gl_v4i;
typedef __attribute__((address_space(3))) v4i lds_v4i;

__device__ __forceinline__ void async_copy_b128(const void* gsrc, void* ldst) {
#if __has_builtin(__builtin_amdgcn_global_load_async_to_lds_b128)
  __builtin_amdgcn_global_load_async_to_lds_b128(
      (gl_v4i*)(uintptr_t)gsrc,
      (lds_v4i*)(unsigned)(uintptr_t)ldst, 0, 0);
#else
  unsigned loff = (unsigned)(uintptr_t)ldst;
  asm volatile("global_load_async_to_lds_b128 %0, %1, off"
               :: "v"(loff), "v"(gsrc) : "memory");
#endif
}

template <int N>
__device__ __forceinline__ void wait_async_le() {
#if __has_builtin(__builtin_amdgcn_s_wait_asynccnt)
  __builtin_amdgcn_s_wait_asynccnt(N);
#else
  asm volatile("s_wait_asynccnt %0" :: "i"(N) : "memory");
#endif
}

__device__ __forceinline__ void wait_ds_zero() {
  asm volatile("s_wait_dscnt 0x0" ::: "memory");
}

// Every thread issues exactly CHUNKS_PER_THREAD async copies (tail indices
// clamp to the last chunk: same-src -> same-dst duplicate writes, benign).
// Uniform issue count makes the "asynccnt <= 10" double-buffer wait exact
// for every wave (async loads complete in order per the ISA).
__device__ __forceinline__ void stage_row_async(const float* __restrict__ row,
                                                float* buf, int tid) {
#pragma unroll
  for (int k = 0; k < CHUNKS_PER_THREAD; ++k) {
    int i = tid + k * BLOCK_THREADS;
    i = (i < VEC16) ? i : (VEC16 - 1);
    async_copy_b128(row + i * 4, buf + i * 4);
  }
}

// Monotone float -> u32 map (ascending order preserved, +inf largest).
__device__ __forceinline__ unsigned order_key32(float f) {
  unsigned b = __float_as_uint(f);
  return b ^ (((unsigned)((int)b >> 31)) | 0x80000000u);
}

__global__ __launch_bounds__(BLOCK_THREADS, 1)
void median_gnn_kernel(const float* __restrict__ x,
                       const float* __restrict__ w,
                       const int*   __restrict__ nidx,
                       const int*   __restrict__ nlen,
                       float* __restrict__ out) {
  __shared__ float stage[2][N_NODES];                                 // 80000 B
  __shared__ unsigned long long keys[2][WAVES_PER_BLOCK][D_MAX];      // 16384 B
  __shared__ float meds[WAVES_PER_BLOCK][2];

  const int tid  = threadIdx.x;
  const int lane = tid & 31;
  const int wv   = tid >> 5;
  const int n    = blockIdx.x * WAVES_PER_BLOCK + wv;   // exact cover of 10000

  const float w0 = w[0], w1 = w[1], w2 = w[2];

  const int len0 = nlen[n];
  const int len1 = nlen[N_NODES + n];
  const int mid0 = (len0 - 1) >> 1;
  const int mid1 = (len1 - 1) >> 1;

  // Lane L owns neighbor slots d = 4L..4L+3; indices stay in VGPRs for all rows.
  const int dbase = lane * 4;
  int4 gi0 = *(const int4*)(nidx + (size_t)n * D_MAX + dbase);
  int4 gi1 = *(const int4*)(nidx + ((size_t)N_NODES + n) * D_MAX + dbase);
  int id0[4] = { gi0.x, gi0.y, gi0.z, gi0.w };
  int id1[4] = { gi1.x, gi1.y, gi1.z, gi1.w };

  unsigned long long* kt0 = keys[0][wv];
  unsigned long long* kt1 = keys[1][wv];
  const ulonglong2* K0 = (const ulonglong2*)kt0;
  const ulonglong2* K1 = (const ulonglong2*)kt1;

  // Prologue: start staging row 0.
  stage_row_async(x, stage[0], tid);

  for (int bf = 0; bf < BF_DIM; ++bf) {
    const int cur = bf & 1;

    // Kick off next row into the other buffer, then wait only for the
    // CURRENT row's 10 in-order copies (the new 10 stay in flight and
    // overlap with this iteration's median compute).
    if (bf + 1 < BF_DIM) {
      stage_row_async(x + (size_t)(bf + 1) * N_NODES, stage[cur ^ 1], tid);
      wait_async_le<CHUNKS_PER_THREAD>();
    } else {
      wait_async_le<0>();
    }
    __syncthreads();   // all waves' slices of the current row are in LDS

    const float* st = stage[cur];

    // ---- gather + build strict-order 64-bit keys, publish to per-wave table ----
    float v0[4], v1[4];
    unsigned long long k0[4], k1[4];
#pragma unroll
    for (int q = 0; q < 4; ++q) {
      const int d = dbase + q;
      float a = st[id0[q]];
      float b = st[id1[q]];
      if (d >= len0) a = __builtin_inff();
      if (d >= len1) b = __builtin_inff();
      v0[q] = a; v1[q] = b;
      k0[q] = ((unsigned long long)order_key32(a) << 32) | (unsigned)d;
      k1[q] = ((unsigned long long)order_key32(b) << 32) | (unsigned)d;
      kt0[d] = k0[q];
      kt1[d] = k1[q];
    }
    wait_ds_zero();   // in-wave: all lanes' key stores visible before reads

    // ---- rank selection: rank = #{keys strictly smaller} (v_cmp_lt_u64) ----
    int r0q[4] = {0, 0, 0, 0};
    int r1q[4] = {0, 0, 0, 0};
#pragma unroll 4
    for (int j2 = 0; j2 < D_MAX / 2; ++j2) {
      ulonglong2 A = K0[j2];
      ulonglong2 B = K1[j2];
#pragma unroll
      for (int q = 0; q < 4; ++q) {
        r0q[q] += (int)(A.x < k0[q]) + (int)(A.y < k0[q]);
        r1q[q] += (int)(B.x < k1[q]) + (int)(B.y < k1[q]);
      }
    }

    // ---- exactly one lane/slot matches the median rank per hop ----
#pragma unroll
    for (int q = 0; q < 4; ++q) {
      if (r0q[q] == mid0) meds[wv][0] = v0[q];
      if (r1q[q] == mid1) meds[wv][1] = v1[q];
    }
    wait_ds_zero();

    if (lane == 0) {
      float xn = st[n];
      out[(size_t)bf * N_NODES + n] = w0 * xn + w1 * meds[wv][0] + w2 * meds[wv][1];
    }
    __syncthreads();   // nobody may restage/overwrite until all finished reading
  }
}

extern "C" void kernel_launch(void* const* d_in, const int* in_sizes, int n_in,
                              void* d_out, int out_size, void* d_ws, size_t ws_size,
                              hipStream_t stream) {
  (void)in_sizes; (void)n_in; (void)out_size; (void)d_ws; (void)ws_size;
  const float* x    = (const float*)d_in[0];   // (2, 64, 10000) f32
  const float* wgt  = (const float*)d_in[1];   // (1, 3) f32
  const int*   nidx = (const int*)d_in[2];     // (2, 10000, 128) i32
  const int*   nlen = (const int*)d_in[3];     // (2, 10000) i32
  float* out = (float*)d_out;                  // (2, 64, 10000) f32

  dim3 grid(N_NODES / WAVES_PER_BLOCK);        // 1250
  dim3 block(BLOCK_THREADS);                   // 256 = 8 waves (wave32)
  hipLaunchKernelGGL(median_gnn_kernel, grid, block, 0, stream,
                     x, wgt, nidx, nlen, out);
}